// FFE_5506148074107
// MI455X (gfx1250) — compile-verified
//
#include <hip/hip_runtime.h>
#include <math.h>

typedef float v2f __attribute__((ext_vector_type(2)));
typedef float v8f __attribute__((ext_vector_type(8)));

#define IMG_H 192
#define IMG_W 192
#define NCH   512
#define NT    4
#define HWSZ  (IMG_H * IMG_W)

// ---------------------------------------------------------------------------
// Kernel A: global average pool per channel.  g[c] = mean(F[c,:,:])
// 512 blocks x 256 threads; float4 loads; LDS tree reduction.
// ---------------------------------------------------------------------------
__global__ void gap_kernel(const float* __restrict__ F, float* __restrict__ g) {
    __shared__ float sdata[256];
    const int c = blockIdx.x;
    const float4* Fc = reinterpret_cast<const float4*>(F + (size_t)c * HWSZ);
    float s = 0.f;
    for (int i = threadIdx.x; i < HWSZ / 4; i += 256) {
        float4 v = Fc[i];
        s += v.x + v.y + v.z + v.w;
    }
    sdata[threadIdx.x] = s;
    __syncthreads();
    for (int off = 128; off > 0; off >>= 1) {
        if ((int)threadIdx.x < off) sdata[threadIdx.x] += sdata[threadIdx.x + off];
        __syncthreads();
    }
    if (threadIdx.x == 0) g[c] = sdata[0] * (1.0f / (float)HWSZ);
}

// ---------------------------------------------------------------------------
// Kernel B: head GEMV via V_WMMA_F32_16X16X4_F32 (one wave), tanh, then emit
//   - output tail: a_exp (16 floats) + B translations (8 floats)
//   - per-t bilinear sampling params into workspace
//
// WMMA mapping: D = A x B + C, A is 16x4 (M=k rows of w1, 8 used), B is 4x16
// with g broadcast along N.  A-layout (ISA 7.12.2, 32-bit A 16x4): lane L,
// VGPR v -> element A[L%16, v + 2*(L/16)].  B 4x16 mirrors C-layout: lane L,
// VGPR v -> B[v + 2*(L/16), L%16].  D: lane 0, VGPR k holds D[k, 0].
//
// A-loads use a clamped row index (m & 7) so they are unconditional (no exec
// masking / branches); the unused rows 8..15 are zeroed with a select that
// lowers to v_cndmask.
// ---------------------------------------------------------------------------
__global__ void head_kernel(const float* __restrict__ g,
                            const float* __restrict__ w1,
                            const float* __restrict__ b1,
                            const float* __restrict__ a,
                            float* __restrict__ params,
                            float* __restrict__ out_tail) {
    const int lane = threadIdx.x;        // 0..31, one full wave (EXEC all ones)
    const int m    = lane & 15;          // M row
    const int kb   = (lane >> 4) * 2;    // K base for this half-wave
    const int rowc = (m & 7) * NCH;      // clamped w1 row (always valid addr)
    const bool live = (m < 8);

    v8f acc = {};
    for (int c0 = 0; c0 < NCH; c0 += 4) {
        // Unconditional loads; zero rows 8..15 via select (v_cndmask).
        float ax = w1[rowc + c0 + kb];
        float ay = w1[rowc + c0 + kb + 1];
        v2f A, B;
        A.x = live ? ax : 0.f;
        A.y = live ? ay : 0.f;
        B.x = g[c0 + kb];
        B.y = g[c0 + kb + 1];
        // v_wmma_f32_16x16x4_f32: D = A*B + acc
        acc = __builtin_amdgcn_wmma_f32_16x16x4_f32(
            /*neg_a=*/false, A, /*neg_b=*/false, B,
            /*c_mod=*/(short)0, acc, /*reuse_a=*/false, /*reuse_b=*/false);
    }

    if (lane == 0) {
        float y[8];
        for (int k = 0; k < 8; ++k) y[k] = tanhf(acc[k] + b1[k]);

        // output tail: a_exp = broadcast of a (identity) to (T,2,2), then B
        for (int t = 0; t < NT; ++t)
            for (int q = 0; q < 4; ++q)
                out_tail[t * 4 + q] = a[q];
        for (int k = 0; k < 8; ++k) out_tail[16 + k] = y[k];

        // Per-t constant shift params: ix = j + tx*W/2, iy = i + ty*H/2
        for (int t = 0; t < NT; ++t) {
            float dx = y[2 * t]     * (0.5f * (float)IMG_W);
            float dy = y[2 * t + 1] * (0.5f * (float)IMG_H);
            float Dx = floorf(dx), Dy = floorf(dy);
            float fx = dx - Dx,    fy = dy - Dy;
            float* p = params + t * 8;
            p[0] = Dx;                     p[1] = Dy;
            p[2] = (1.f - fy) * (1.f - fx);
            p[3] = (1.f - fy) * fx;
            p[4] = fy * (1.f - fx);
            p[5] = fy * fx;
        }
    }
}

// ---------------------------------------------------------------------------
// Kernel C: uniform sub-pixel bilinear shift of F, zero padding.
// One block per (t, c) plane; 192 threads = one per column (coalesced);
// loop over 192 rows.  Column validity hoisted out of the loop.
// ---------------------------------------------------------------------------
__global__ void shift_kernel(const float* __restrict__ F,
                             const float* __restrict__ params,
                             float* __restrict__ out) {
    const int bc = blockIdx.x;           // 0 .. NT*NCH-1
    const int t  = bc >> 9;              // / 512
    const int c  = bc & 511;

    const float* p = params + t * 8;
    const int   Dx  = (int)p[0];
    const int   Dy  = (int)p[1];
    const float w00 = p[2], w01 = p[3], w10 = p[4], w11 = p[5];

    const float* Fc = F + (size_t)c * HWSZ;
    float* Oc = out + ((size_t)t * NCH + c) * HWSZ;

    const int j  = threadIdx.x;          // 0..191
    const int x0 = j + Dx;
    const int x1 = x0 + 1;
    const bool c0ok = (x0 >= 0) & (x0 < IMG_W);
    const bool c1ok = (x1 >= 0) & (x1 < IMG_W);

    for (int i = 0; i < IMG_H; ++i) {
        const int y0 = i + Dy, y1 = y0 + 1;
        const bool r0 = (y0 >= 0) & (y0 < IMG_H);
        const bool r1 = (y1 >= 0) & (y1 < IMG_H);
        const float* row0 = Fc + y0 * IMG_W;
        const float* row1 = Fc + y1 * IMG_W;
        float g00 = (r0 & c0ok) ? row0[x0] : 0.f;
        float g01 = (r0 & c1ok) ? row0[x1] : 0.f;
        float g10 = (r1 & c0ok) ? row1[x0] : 0.f;
        float g11 = (r1 & c1ok) ? row1[x1] : 0.f;
        Oc[i * IMG_W + j] = w00 * g00 + w01 * g01 + w10 * g10 + w11 * g11;
    }
}

// ---------------------------------------------------------------------------
// Launch: inputs = {F, w1, b1, a}; output = trans_x (T*512*192*192) | a_exp
// (16) | B (8).  Workspace: g (512 floats) + params (32 floats).
// ---------------------------------------------------------------------------
extern "C" void kernel_launch(void* const* d_in, const int* in_sizes, int n_in,
                              void* d_out, int out_size, void* d_ws, size_t ws_size,
                              hipStream_t stream) {
    (void)in_sizes; (void)n_in; (void)out_size; (void)ws_size;

    const float* F  = (const float*)d_in[0];
    const float* w1 = (const float*)d_in[1];
    const float* b1 = (const float*)d_in[2];
    const float* a  = (const float*)d_in[3];

    float* ws     = (float*)d_ws;
    float* g      = ws;            // 512 floats
    float* params = ws + 512;      // 32 floats

    float* out      = (float*)d_out;
    float* out_tail = out + (size_t)NT * NCH * HWSZ;

    gap_kernel<<<NCH, 256, 0, stream>>>(F, g);
    head_kernel<<<1, 32, 0, stream>>>(g, w1, b1, a, params, out_tail);
    shift_kernel<<<NT * NCH, IMG_W, 0, stream>>>(F, params, out);
}